// RegularizedFMNet_3788161155487
// MI455X (gfx1250) — compile-verified
//
#include <hip/hip_runtime.h>
#include <hip/hip_bf16.h>
#include <stdint.h>

// ---------------------------------------------------------------------------
// RegularizedFMNet on MI455X (gfx1250, wave32, WMMA, async global->LDS DMA)
//   1) gemm_proj:  A = Ex(KxV) x Fx(VxC), Bm = Ey x Fy
//        - v_wmma_f32_16x16x4_f32 (fp32 in/out: feeds a linear solve)
//        - V-chunks of 20 staged into LDS via global_load_async_to_lds_b128,
//          double buffered, s_wait_asynccnt-pipelined (5 issues/wave/chunk)
//        - V split 4 ways -> deterministic partials in workspace
//   2) reduce_partials: sum the 4 V-splits
//   3) mask_kernel: resolvant mask D (tiny)
//   4) gram_kernel: A_A_t, B_A_t (L2-resident dots)
//   5) solve_kernel: 1024 independent 128x128 SPD solves, Gaussian elim in
//      dynamic LDS (128x129 padding -> conflict-free row sweeps)
// ---------------------------------------------------------------------------

#define B_ 8
#define V_ 30000
#define C_ 256
#define K_ 128
#define SPLIT_ 4
#define VSPLIT_ (V_ / SPLIT_)          // 7500 v per split
#define CHUNKV_ 20                     // v-depth staged per chunk (5 wmma steps)
#define NCHUNKS_ (VSPLIT_ / CHUNKV_)   // 375
#define LMBDA_ 100.0f
#define LDSTRIDE_ 129

// LDS float offsets inside the gemm workgroup allocation
#define SA_FLOATS_ (K_ * CHUNKV_)          // 2560 floats per A buffer
#define SF_FLOATS_ (CHUNKV_ * 64)          // 1280 floats per F buffer
#define SF_BASE_   (2 * SA_FLOATS_)        // 5120
#define GEMM_LDS_BYTES_ ((2 * SA_FLOATS_ + 2 * SF_FLOATS_) * 4)  // 30720 B

typedef __attribute__((ext_vector_type(2))) float v2f;
typedef __attribute__((ext_vector_type(8))) float v8f;

// Async DMA: global (b128) -> LDS. Tracked by ASYNCcnt.
__device__ __forceinline__ void async_copy_b128(uint32_t lds_byte_off,
                                                const float* gptr)
{
    uint64_t ga = (uint64_t)(uintptr_t)gptr;
    asm volatile("global_load_async_to_lds_b128 %0, %1, off"
                 :: "v"(lds_byte_off), "v"(ga) : "memory");
}

// Issue one chunk's staging: E tile 128x20 (640 b128), F tile 20x64 (320 b128).
// Exactly 5 async instructions per wave (wave-uniform -> fixed asynccnt math).
__device__ __forceinline__ void issue_chunk(
    const float* __restrict__ E0,   // E + b*K*V        (row stride V)
    const float* __restrict__ F0,   // F + b*V*C + col0 (row stride C)
    uint32_t ldsA, uint32_t ldsF,   // byte offsets of target buffers
    int v0, int wave, int lane)
{
    // --- E tile: 640 b128 segments, 80 per wave (rows of 5 segments) ---
    int idx = wave * 80 + lane;
    {
        int row = idx / 5, seg = idx - row * 5;
        async_copy_b128(ldsA + (uint32_t)(row * CHUNKV_ + seg * 4) * 4u,
                        E0 + (size_t)row * V_ + v0 + seg * 4);
    }
    idx += 32;
    {
        int row = idx / 5, seg = idx - row * 5;
        async_copy_b128(ldsA + (uint32_t)(row * CHUNKV_ + seg * 4) * 4u,
                        E0 + (size_t)row * V_ + v0 + seg * 4);
    }
    idx += 32;
    if (lane < 16) {   // partial EXEC, still one wave-issue
        int row = idx / 5, seg = idx - row * 5;
        async_copy_b128(ldsA + (uint32_t)(row * CHUNKV_ + seg * 4) * 4u,
                        E0 + (size_t)row * V_ + v0 + seg * 4);
    }
    // --- F tile: 320 b128 segments, 40 per wave (rows of 16 segments) ---
    int fidx = wave * 40 + lane;
    {
        int row = fidx >> 4, seg = fidx & 15;
        async_copy_b128(ldsF + (uint32_t)(row * 64 + seg * 4) * 4u,
                        F0 + (size_t)(v0 + row) * C_ + seg * 4);
    }
    if (lane < 8) {    // partial EXEC, still one wave-issue
        int r2 = (fidx + 32) >> 4, s2 = (fidx + 32) & 15;
        async_copy_b128(ldsF + (uint32_t)(r2 * 64 + s2 * 4) * 4u,
                        F0 + (size_t)(v0 + r2) * C_ + s2 * 4);
    }
}

// ---------------------------------------------------------------------------
// Kernel 1: fp32 WMMA GEMM with async-LDS double buffering.
// grid = (C/64, 2*B, SPLIT), block = 256 (8 waves); wave tile = 32x32.
// ---------------------------------------------------------------------------
__global__ void __launch_bounds__(256) gemm_proj(
    const float* __restrict__ fx, const float* __restrict__ fy,
    const float* __restrict__ ex, const float* __restrict__ ey,
    float* __restrict__ part)
{
    extern __shared__ float smem[];   // [2][128][20] A  +  [2][20][64] F

    const int matb = blockIdx.y;      // 2*b + mat
    const int mat  = matb & 1;
    const int b    = matb >> 1;
    const float* __restrict__ F = mat ? fy : fx;   // V x C
    const float* __restrict__ E = mat ? ey : ex;   // K x V

    const int wave  = threadIdx.x >> 5;
    const int lane  = threadIdx.x & 31;
    const int half  = lane >> 4;       // A frag: 0 -> K pair {0,1}, 1 -> {2,3}
    const int lanem = lane & 15;

    const int row_base = (wave >> 1) * 32;   // 0,32,64,96
    const int colh     = (wave & 1) * 32;    // 0 or 32 within the 64-col tile

    const float* __restrict__ E0 = E + (size_t)b * K_ * V_;
    const float* __restrict__ F0 = F + (size_t)b * V_ * C_ + blockIdx.x * 64;

    const uint32_t ldsbase = (uint32_t)(uintptr_t)smem;
    const int v0 = blockIdx.z * VSPLIT_;

    v8f acc00 = {}, acc01 = {}, acc10 = {}, acc11 = {};

    // Prologue: stage chunk 0 into buffer 0.
    issue_chunk(E0, F0, ldsbase, ldsbase + SF_BASE_ * 4u, v0, wave, lane);

    for (int c = 0; c < NCHUNKS_; ++c) {
        const int buf = c & 1;
        if (c + 1 < NCHUNKS_) {
            const int nb = buf ^ 1;
            issue_chunk(E0, F0,
                        ldsbase + (uint32_t)(nb * SA_FLOATS_) * 4u,
                        ldsbase + (uint32_t)(SF_BASE_ + nb * SF_FLOATS_) * 4u,
                        v0 + (c + 1) * CHUNKV_, wave, lane);
            // 5 in-flight issues belong to chunk c+1; <=5 means chunk c landed
            asm volatile("s_wait_asynccnt 0x5" ::: "memory");
        } else {
            asm volatile("s_wait_asynccnt 0x0" ::: "memory");
        }
        __syncthreads();   // chunk c visible to every wave

        const float* __restrict__ As = smem + buf * SA_FLOATS_;
        const float* __restrict__ Fs = smem + SF_BASE_ + buf * SF_FLOATS_;

#pragma unroll
        for (int st = 0; st < CHUNKV_ / 4; ++st) {
            const int vv = st * 4 + 2 * half;
            // A fragments (ds_load_b64, conflict-free: stride 20 dwords)
            v2f a0 = *(const v2f*)(As + (row_base + lanem) * CHUNKV_ + vv);
            v2f a1 = *(const v2f*)(As + (row_base + 16 + lanem) * CHUNKV_ + vv);
            // B fragments (consecutive-bank ds_load_b32)
            const float* fr = Fs + vv * 64 + colh + lanem;
            v2f b0, b1;
            b0.x = fr[0];  b0.y = fr[64];
            b1.x = fr[16]; b1.y = fr[80];

            acc00 = __builtin_amdgcn_wmma_f32_16x16x4_f32(false, a0, false, b0,
                                                          (short)0, acc00, false, false);
            acc01 = __builtin_amdgcn_wmma_f32_16x16x4_f32(false, a0, false, b1,
                                                          (short)0, acc01, false, false);
            acc10 = __builtin_amdgcn_wmma_f32_16x16x4_f32(false, a1, false, b0,
                                                          (short)0, acc10, false, false);
            acc11 = __builtin_amdgcn_wmma_f32_16x16x4_f32(false, a1, false, b1,
                                                          (short)0, acc11, false, false);
        }
        __syncthreads();   // everyone done reading buf before it is re-filled
    }

    // C/D layout: VGPR r <-> row (half*8 + r), col = lanem
    float* __restrict__ P =
        part + ((size_t)blockIdx.z * (2 * B_) + matb) * (size_t)(K_ * C_);
    const int colw = blockIdx.x * 64 + colh + lanem;
#pragma unroll
    for (int r = 0; r < 8; ++r) {
        const int rowA = row_base + half * 8 + r;
        const int rowB = rowA + 16;
        P[(size_t)rowA * C_ + colw]      = acc00[r];
        P[(size_t)rowA * C_ + colw + 16] = acc01[r];
        P[(size_t)rowB * C_ + colw]      = acc10[r];
        P[(size_t)rowB * C_ + colw + 16] = acc11[r];
    }
}

// ---------------------------------------------------------------------------
// Kernel 2: reduce the SPLIT_ partial sums. AB layout: [2b+mat][K][C].
// ---------------------------------------------------------------------------
__global__ void reduce_partials(const float* __restrict__ part,
                                float* __restrict__ AB, int total)
{
    int t = blockIdx.x * blockDim.x + threadIdx.x;
    if (t < total) {
        float s = 0.0f;
#pragma unroll
        for (int i = 0; i < SPLIT_; ++i) s += part[(size_t)i * total + t];
        AB[t] = s;
    }
}

// ---------------------------------------------------------------------------
// Kernel 3: resolvant mask D[b][i][j]. grid = B, block = 128 (thread = j).
// ---------------------------------------------------------------------------
__global__ void mask_kernel(const float* __restrict__ evx,
                            const float* __restrict__ evy,
                            float* __restrict__ Dm)
{
    const int b = blockIdx.x, t = threadIdx.x;
    __shared__ float sx[K_], sy[K_], red[K_];

    const float vx = evx[b * K_ + t];
    const float vy = evy[b * K_ + t];
    red[t] = fmaxf(vx, vy);
    __syncthreads();
    for (int off = 64; off > 0; off >>= 1) {
        if (t < off) red[t] = fmaxf(red[t], red[t + off]);
        __syncthreads();
    }
    const float scal = red[0];
    sx[t] = sqrtf(vx / scal);   // gamma = 0.5, evals >= 0
    sy[t] = sqrtf(vy / scal);
    __syncthreads();

    const float g2 = sy[t];
    const float d2 = 1.0f / (g2 * g2 + 1.0f);
    for (int i = 0; i < K_; ++i) {
        const float g1 = sx[i];
        const float d1 = 1.0f / (g1 * g1 + 1.0f);
        const float re = g2 * d2 - g1 * d1;
        const float im = d2 - d1;
        Dm[((size_t)b * K_ + i) * K_ + t] = re * re + im * im;
    }
}

// ---------------------------------------------------------------------------
// Kernel 4: Gram matrices. grid = (K, B, 2), block = 128 (thread = l).
// ---------------------------------------------------------------------------
__global__ void gram_kernel(const float* __restrict__ AB,
                            float* __restrict__ AAt,
                            float* __restrict__ BAt)
{
    const int k = blockIdx.x, b = blockIdx.y, which = blockIdx.z;
    const int l = threadIdx.x;
    __shared__ float srow[C_];

    const float* __restrict__ A_mat = AB + (size_t)(2 * b) * K_ * C_;
    const float* __restrict__ src =
        which ? AB + (size_t)(2 * b + 1) * K_ * C_ + (size_t)k * C_
              : A_mat + (size_t)k * C_;
    srow[l]       = src[l];
    srow[l + 128] = src[l + 128];
    __syncthreads();

    const float* __restrict__ al = A_mat + (size_t)l * C_;
    float acc = 0.0f;
#pragma unroll 4
    for (int c = 0; c < C_; ++c) acc += srow[c] * al[c];

    float* dst = which ? BAt : AAt;
    dst[((size_t)b * K_ + k) * K_ + l] = acc;
}

// ---------------------------------------------------------------------------
// Kernel 5: 1024 independent SPD solves in dynamic LDS. grid = B*K, block=128.
// ---------------------------------------------------------------------------
__global__ void __launch_bounds__(128) solve_kernel(
    const float* __restrict__ AAt, const float* __restrict__ BAt,
    const float* __restrict__ Dm, float* __restrict__ out)
{
    const int bm = blockIdx.x;
    const int b  = bm >> 7;
    const int m  = bm & (K_ - 1);
    const int i  = threadIdx.x;

    extern __shared__ float sm[];
    float* M   = sm;                       // K_ * LDSTRIDE_
    float* rhs = sm + K_ * LDSTRIDE_;      // K_
    float* xv  = rhs + K_;                 // K_

    const float* __restrict__ Ab = AAt + (size_t)b * K_ * K_;
    for (int c = 0; c < K_; ++c) M[i * LDSTRIDE_ + c] = Ab[(size_t)i * K_ + c];
    M[i * LDSTRIDE_ + i] += LMBDA_ * Dm[((size_t)b * K_ + m) * K_ + i];
    rhs[i] = BAt[((size_t)b * K_ + m) * K_ + i];
    __syncthreads();

    for (int j = 0; j < K_; ++j) {
        if (i > j) {
            const float f = M[i * LDSTRIDE_ + j] / M[j * LDSTRIDE_ + j];
            for (int c = j + 1; c < K_; ++c)
                M[i * LDSTRIDE_ + c] -= f * M[j * LDSTRIDE_ + c];
            rhs[i] -= f * rhs[j];
        }
        __syncthreads();
    }

    for (int j = K_ - 1; j >= 0; --j) {
        if (i == j) xv[j] = rhs[j] / M[j * LDSTRIDE_ + j];
        __syncthreads();
        if (i < j) rhs[i] -= M[i * LDSTRIDE_ + j] * xv[j];
        __syncthreads();
    }

    out[((size_t)b * K_ + m) * K_ + i] = xv[i];
}

// ---------------------------------------------------------------------------
// Launch. Workspace layout (floats):
//   partials : SPLIT*2*B*K*C = 2,097,152
//   AB       : 2*B*K*C       =   524,288
//   D        : B*K*K         =   131,072
//   A_A_t    : B*K*K         =   131,072
//   B_A_t    : B*K*K         =   131,072      total ~11.5 MB
// ---------------------------------------------------------------------------
extern "C" void kernel_launch(void* const* d_in, const int* in_sizes, int n_in,
                              void* d_out, int out_size, void* d_ws, size_t ws_size,
                              hipStream_t stream)
{
    const float* feat_x  = (const float*)d_in[0];
    const float* feat_y  = (const float*)d_in[1];
    const float* evals_x = (const float*)d_in[2];
    const float* evals_y = (const float*)d_in[3];
    const float* etx     = (const float*)d_in[4];
    const float* ety     = (const float*)d_in[5];
    float* outp = (float*)d_out;

    float* ws   = (float*)d_ws;
    float* part = ws;
    float* AB   = part + (size_t)SPLIT_ * 2 * B_ * K_ * C_;
    float* Dm   = AB   + (size_t)2 * B_ * K_ * C_;
    float* AAt  = Dm   + (size_t)B_ * K_ * K_;
    float* BAt  = AAt  + (size_t)B_ * K_ * K_;

    dim3 g1(C_ / 64, 2 * B_, SPLIT_);
    gemm_proj<<<g1, 256, GEMM_LDS_BYTES_, stream>>>(feat_x, feat_y, etx, ety, part);

    const int total = 2 * B_ * K_ * C_;
    reduce_partials<<<(total + 255) / 256, 256, 0, stream>>>(part, AB, total);

    mask_kernel<<<B_, K_, 0, stream>>>(evals_x, evals_y, Dm);

    dim3 g3(K_, B_, 2);
    gram_kernel<<<g3, K_, 0, stream>>>(AB, AAt, BAt);

    const size_t shsz = (size_t)(K_ * LDSTRIDE_ + 2 * K_) * sizeof(float);
    solve_kernel<<<B_ * K_, K_, shsz, stream>>>(AAt, BAt, Dm, outp);
}